// YOLOV3Loss_82884278879270
// MI455X (gfx1250) — compile-verified
//
#include <hip/hip_runtime.h>
#include <math.h>

#define TGT  20
#define NCLS 80

typedef float v2f __attribute__((ext_vector_type(2)));
typedef float v8f __attribute__((ext_vector_type(8)));

// BCE on probabilities with log clamped at -100, expressed on logits:
// bce(sigmoid(z), t) = t*min(softplus(-z),100) + (1-t)*min(softplus(z),100)
__device__ __forceinline__ float softplus_f(float z) {
    float a = fabsf(z);
    float r = log1pf(expf(-a));
    return z > 0.f ? z + r : r;
}
__device__ __forceinline__ float bce_z(float z, float t) {
    float spn = fminf(softplus_f(-z), 100.f);
    float spp = fminf(softplus_f(z), 100.f);
    return t * spn + (1.f - t) * spp;
}

// Sum one f32 value across all 32 lanes of a wave using the f32 WMMA
// (ones-matrix reduction; exact IEEE f32). A-matrix 16x4 layout: lane L<16
// holds A[L][0..1] in v0..v1, lane L>=16 holds A[L-16][2..3]. With a=(v,0)
// and B=ones, D[m][n] = v_m + v_{m+16}; each lane sums its 8 D rows, then
// one xor-16 shuffle completes the 32-lane sum. EXEC must be all 1s: callers
// guarantee full waves (no early returns).
__device__ __forceinline__ float wave_sum32(float v) {
    v2f a; a[0] = v;   a[1] = 0.f;
    v2f b; b[0] = 1.f; b[1] = 1.f;
    v8f c = {};
    v8f d = __builtin_amdgcn_wmma_f32_16x16x4_f32(
        /*neg_a=*/false, a, /*neg_b=*/false, b,
        /*c_mod=*/(short)0, c, /*reuse_a=*/false, /*reuse_b=*/false);
    float s = 0.f;
#pragma unroll
    for (int k = 0; k < 8; ++k) s += d[k];
    s += __shfl_xor(s, 16, 32);
    return s;
}

__global__ __launch_bounds__(256) void yolo_level_kernel(
    const float* __restrict__ x,     // [bs,255,H,W]
    const float* __restrict__ gtb,   // [bs,TGT,4] xyxy pixels, -1 = pad
    const int*   __restrict__ glab,  // [bs,TGT]
    float*       __restrict__ partial,
    int H, int W, float stride,
    float sa0x, float sa0y, float sa1x, float sa1y, float sa2x, float sa2y)
{
    const int HW    = H * W;
    const int cells = 3 * HW;
    const int b     = blockIdx.y;
    const int tid   = threadIdx.x;

    __shared__ float tx1[TGT], ty1[TGT], tx2[TGT], ty2[TGT], tarea[TGT];
    __shared__ float ttx[TGT], tty[TGT], ttw[TGT], tth[TGT];
    __shared__ int   tflat[TGT], tlab[TGT], tval[TGT];
    __shared__ float swave[8];

    if (tid < TGT) {
        const float* g = gtb + ((size_t)b * TGT + tid) * 4;
        float X1 = g[0], Y1 = g[1], X2 = g[2], Y2 = g[3];
        int valid = (X1 != -1.f) || (Y1 != -1.f) || (X2 != -1.f) || (Y2 != -1.f);
        float inv = 1.f / stride;   // stride is a power of two -> exact
        float gw  = (X2 - X1) * inv, gh = (Y2 - Y1) * inv;
        float gxc = (X1 + X2) * (0.5f * inv), gyc = (Y1 + Y2) * (0.5f * inv);
        int gi = (int)gxc; gi = gi < 0 ? 0 : (gi > W - 1 ? W - 1 : gi);
        int gj = (int)gyc; gj = gj < 0 ? 0 : (gj > H - 1 ? H - 1 : gj);
        float ax[3] = { sa0x, sa1x, sa2x };
        float ay[3] = { sa0y, sa1y, sa2y };
        float best = -1.f; int bn = 0;
#pragma unroll
        for (int a = 0; a < 3; ++a) {
            float inter = fminf(gw, ax[a]) * fminf(gh, ay[a]);
            float iou   = inter / (gw * gh + ax[a] * ay[a] - inter + 1e-16f);
            if (iou > best) { best = iou; bn = a; }   // first max wins (argmax)
        }
        tflat[tid] = bn * HW + gj * W + gi;
        float c0 = X1 * inv, c1 = Y1 * inv, c2 = X2 * inv, c3 = Y2 * inv;
        tx1[tid] = c0; ty1[tid] = c1; tx2[tid] = c2; ty2[tid] = c3;
        tarea[tid] = (c2 - c0) * (c3 - c1);
        ttx[tid] = gxc - (float)gi;
        tty[tid] = gyc - (float)gj;
        ttw[tid] = logf(gw / ax[bn] + 1e-16f);
        tth[tid] = logf(gh / ay[bn] + 1e-16f);
        int lb = glab[b * TGT + tid];
        tlab[tid] = lb < 0 ? 0 : (lb > NCLS - 1 ? NCLS - 1 : lb);
        tval[tid] = valid;
    }
    __syncthreads();

    int cell = blockIdx.x * blockDim.x + tid;
    float contrib = 0.f;
    if (cell < cells) {
        int a   = cell / HW;
        int rem = cell - a * HW;
        int j   = rem / W;
        int i   = rem - j * W;
        const float* xb = x + ((size_t)b * 255 + (size_t)a * 85) * HW + rem;
        float z0 = xb[0], z1 = xb[HW], z2 = xb[2 * HW], z3 = xb[3 * HW], z4 = xb[4 * HW];
        float sax = (a == 0) ? sa0x : ((a == 1) ? sa1x : sa2x);
        float say = (a == 0) ? sa0y : ((a == 1) ? sa1y : sa2y);
        float px = 1.f / (1.f + expf(-z0));
        float py = 1.f / (1.f + expf(-z1));
        float bx = px + (float)i, by = py + (float)j;
        float bw = expf(z2) * sax, bh = expf(z3) * say;
        float pbx1 = bx - 0.5f * bw, pby1 = by - 0.5f * bh;
        float pbx2 = bx + 0.5f * bw, pby2 = by + 0.5f * bh;
        float areap = (pbx2 - pbx1) * (pby2 - pby1);

        int Lig = -1, Lpos = -1;
        unsigned cb0 = 0, cb1 = 0, cb2 = 0;
#pragma unroll
        for (int t = 0; t < TGT; ++t) {
            if (!tval[t]) continue;
            float ix1 = fmaxf(pbx1, tx1[t]), iy1 = fmaxf(pby1, ty1[t]);
            float ix2 = fminf(pbx2, tx2[t]), iy2 = fminf(pby2, ty2[t]);
            float inter = fmaxf(ix2 - ix1, 0.f) * fmaxf(iy2 - iy1, 0.f);
            float iou   = inter / (areap + tarea[t] - inter + 1e-16f);
            if (iou > 0.5f) Lig = t;                 // last ignoring target
            if (tflat[t] == cell) {                  // last positive writer
                Lpos = t;
                int lb = tlab[t];                    // tcls: every hit sets its label
                if (lb < 32)      cb0 |= 1u << lb;
                else if (lb < 64) cb1 |= 1u << (lb - 32);
                else              cb2 |= 1u << (lb - 64);
            }
        }
        bool pos = (Lpos >= 0) && (Lpos >= Lig);     // positives win ties
        bool neg = (!pos) && (Lig < 0);
        if (pos) {
            int wn = Lpos;
            contrib += 2.5f * (bce_z(z0, ttx[wn]) + bce_z(z1, tty[wn]));
            float dw = z2 - ttw[wn], dh = z3 - tth[wn];
            contrib += 2.5f * (dw * dw + dh * dh);
            contrib += fminf(softplus_f(-z4), 100.f);            // bce(conf, 1)
            for (int c = 0; c < NCLS; ++c) {                     // rare path
                float zc = xb[(size_t)(5 + c) * HW];
                unsigned bit = (c < 32) ? ((cb0 >> c) & 1u)
                             : (c < 64) ? ((cb1 >> (c - 32)) & 1u)
                                        : ((cb2 >> (c - 64)) & 1u);
                contrib += bce_z(zc, (float)bit);
            }
        } else if (neg) {
            contrib += fminf(softplus_f(z4), 100.f);             // bce(conf, 0)
        }
        // maskf == -1 (ignored) contributes nothing
    }

    // Block reduction: WMMA ones-reduction per wave (full EXEC), then LDS.
    float ws = wave_sum32(contrib);
    int wave = tid >> 5, lane = tid & 31;
    if (lane == 0) swave[wave] = ws;
    __syncthreads();
    if (tid == 0) {
        float s = 0.f;
#pragma unroll
        for (int k = 0; k < 8; ++k) s += swave[k];
        partial[(size_t)blockIdx.y * gridDim.x + blockIdx.x] = s;
    }
}

__global__ __launch_bounds__(32) void yolo_reduce_kernel(
    const float* __restrict__ partial, int n, float* __restrict__ out, float scale)
{
    int lane = threadIdx.x;                 // single wave, full EXEC
    float acc = 0.f;
    for (int k = lane; k < n; k += 32) acc += partial[k];  // fixed order: deterministic
    float total = wave_sum32(acc);
    if (lane == 0) out[0] = total * scale;
}

extern "C" void kernel_launch(void* const* d_in, const int* in_sizes, int n_in,
                              void* d_out, int out_size, void* d_ws, size_t ws_size,
                              hipStream_t stream) {
    (void)in_sizes; (void)n_in; (void)out_size; (void)ws_size;
    const float* x0   = (const float*)d_in[0];   // [32,255,13,13]
    const float* x1   = (const float*)d_in[1];   // [32,255,26,26]
    const float* x2   = (const float*)d_in[2];   // [32,255,52,52]
    const float* gtb  = (const float*)d_in[3];   // [32,20,4]
    const int*   glab = (const int*)  d_in[4];   // [32,20]
    float* part = (float*)d_ws;
    const int bs = 32;

    struct Lv { const float* x; int H; float stride; float sa[6]; int blocks; };
    Lv lv[3] = {
        { x0, 13, 32.f, {116.f/32.f, 90.f/32.f, 156.f/32.f, 198.f/32.f, 373.f/32.f, 326.f/32.f},
          (3*13*13 + 255) / 256 },
        { x1, 26, 16.f, { 30.f/16.f, 61.f/16.f,  62.f/16.f,  45.f/16.f,  59.f/16.f, 119.f/16.f},
          (3*26*26 + 255) / 256 },
        { x2, 52,  8.f, { 10.f/8.f,  13.f/8.f,   16.f/8.f,   30.f/8.f,   33.f/8.f,  23.f/8.f},
          (3*52*52 + 255) / 256 },
    };

    int off = 0;
    for (int l = 0; l < 3; ++l) {
        dim3 grid(lv[l].blocks, bs);
        yolo_level_kernel<<<grid, 256, 0, stream>>>(
            lv[l].x, gtb, glab, part + off, lv[l].H, lv[l].H, lv[l].stride,
            lv[l].sa[0], lv[l].sa[1], lv[l].sa[2], lv[l].sa[3], lv[l].sa[4], lv[l].sa[5]);
        off += lv[l].blocks * bs;
    }
    yolo_reduce_kernel<<<1, 32, 0, stream>>>(part, off, (float*)d_out, 1.f / (float)bs);
}